// UNet5LMCD_6451040878761
// MI455X (gfx1250) — compile-verified
//
#include <hip/hip_runtime.h>
#include <hip/hip_bf16.h>

typedef __attribute__((ext_vector_type(2))) float v2f;
typedef __attribute__((ext_vector_type(8))) float v8f;

// ---------------------------------------------------------------------------
// Implicit-GEMM conv layer using V_WMMA_F32_16X16X4_F32 (exact fp32 math).
// Fully templated: grid sides are powers of two (mask/shift addressing),
// the K loop over channels fully unrolls into back-to-back v_wmma ops.
//   M = 32 output voxels per block (two 16-row WMMA tiles -> two accumulators,
//       B fragment reused twice), N = 16 cols per wave, K = 27*CIN.
//   MODE: 0 = forward stride 1, 1 = forward stride 2, 2 = transposed stride 2
//   Epilogue: optional BN(eval)+ReLU, then output-resolution sparsity mask.
//   Output written with channel stride/offset so concat buffers are built
//   in place.
// ---------------------------------------------------------------------------
template <int CIN, int COUT, int DIN, int DOUT, int MODE, int DOBN>
__global__ __launch_bounds__(256) void conv_wmma_kernel(
    const float* __restrict__ in, const float* __restrict__ w,
    const float* __restrict__ bn, const float* __restrict__ maskOut,
    float* __restrict__ out, int outStride, int outOffset)
{
    constexpr int CIN4  = CIN / 4;                 // all CIN are multiples of 4
    constexpr int LOG   = (DOUT == 64) ? 6 : (DOUT == 32) ? 5
                        : (DOUT == 16) ? 4 : (DOUT == 8) ? 3 : 2;
    constexpr bool NTAIL = (COUT & 15) != 0;       // Cout = 8 / 2 layers

    __shared__ float4 sA4[32 * CIN4];              // 32 voxels x CIN floats
    __shared__ int    sIn[32];                     // per-tap input voxel (-1 = 0)
    float* sA = (float*)sA4;

    const int tid      = threadIdx.x;
    const int lane     = tid & 31;
    const int nthreads = blockDim.x;
    const int coutBase = (tid >> 5) << 4;          // 16-wide N tile per wave
    const int tileBase = blockIdx.x << 5;          // 32 output voxels per block

    // CDNA5 wave32 fragment coordinates (ISA 7.12.2):
    //  A 16x4 : lanes 0-15 hold M=lane, K={0,1}; lanes 16-31 hold K={2,3}
    //  B 4x16 : lanes 0-15 hold N=lane, K={0,1}; lanes 16-31 hold K={2,3}
    //  C 16x16: VGPR r -> M=r (lanes 0-15) / M=r+8 (lanes 16-31), N=lane&15
    const int rowA = lane & 15;
    const int krow = (lane >> 4) << 1;             // 0 or 2
    const int col  = lane & 15;
    const int coB  = coutBase + col;

    v8f acc0 = {0.f, 0.f, 0.f, 0.f, 0.f, 0.f, 0.f, 0.f};
    v8f acc1 = {0.f, 0.f, 0.f, 0.f, 0.f, 0.f, 0.f, 0.f};

    for (int t = 0; t < 27; ++t) {
        const int tz = t / 9;
        const int ty = (t / 3) % 3;
        const int tx = t % 3;

        __syncthreads();   // protect sA/sIn from previous-tap readers
        if (tid < 32) {
            const int v = tileBase + tid;
            const int x = v & (DOUT - 1);
            const int y = (v >> LOG) & (DOUT - 1);
            const int z = v >> (2 * LOG);
            int ivox = -1;
            if (MODE == 2) {
                // convT stride 2, pad (1,0): in = (o + t - 1)/2 when even
                const int nx = x + tx - 1, ny = y + ty - 1, nz = z + tz - 1;
                if (!((nx | ny | nz) & 1)) {       // all even (and >= 0)
                    const int ix = nx >> 1, iy = ny >> 1, iz = nz >> 1;
                    if (ix < DIN && iy < DIN && iz < DIN)
                        ivox = (iz * DIN + iy) * DIN + ix;
                }
            } else {
                constexpr int S = (MODE == 1) ? 2 : 1;
                const int ix = x * S + tx - 1;
                const int iy = y * S + ty - 1;
                const int iz = z * S + tz - 1;
                if (ix >= 0 && ix < DIN && iy >= 0 && iy < DIN &&
                    iz >= 0 && iz < DIN)
                    ivox = (iz * DIN + iy) * DIN + ix;
            }
            sIn[tid] = ivox;
        }
        __syncthreads();

        // b128 cooperative gather of the 32 x CIN activation slab
        // (NDHWC: channels contiguous, rows 16B-aligned since CIN % 4 == 0).
        for (int idx = tid; idx < 32 * CIN4; idx += nthreads) {
            const int r  = idx / CIN4;
            const int c4 = idx - r * CIN4;
            const int iv = sIn[r];
            float4 val = make_float4(0.f, 0.f, 0.f, 0.f);
            if (iv >= 0)
                val = ((const float4*)(in + (size_t)iv * CIN))[c4];
            sA4[idx] = val;
        }

        // prefetch next tap's weight slab (global_prefetch_b8; L2-resident)
        const float* wt = w + (size_t)t * CIN * COUT;
        if (t < 26) {
            const int pidx = tid * 16;             // one 64B line per thread
            if (pidx < CIN * COUT)
                __builtin_prefetch(wt + CIN * COUT + pidx, 0, 1);
        }
        __syncthreads();

        // K loop fully unrolled: CIN/4 pairs of back-to-back v_wmma.
#pragma unroll
        for (int k = 0; k < CIN; k += 4) {
            v2f a0, a1, b;
            a0.x = sA[rowA * CIN + k + krow];
            a0.y = sA[rowA * CIN + k + krow + 1];
            a1.x = sA[(rowA + 16) * CIN + k + krow];
            a1.y = sA[(rowA + 16) * CIN + k + krow + 1];
            float b0 = 0.0f, b1 = 0.0f;
            if (!NTAIL || coB < COUT) {
                b0 = wt[(k + krow) * COUT + coB];
                b1 = wt[(k + krow + 1) * COUT + coB];
            }
            b.x = b0;
            b.y = b1;
            acc0 = __builtin_amdgcn_wmma_f32_16x16x4_f32(
                false, a0, false, b, (short)0, acc0, false, false);
            acc1 = __builtin_amdgcn_wmma_f32_16x16x4_f32(
                false, a1, false, b, (short)0, acc1, false, false);
        }
    }

    // Epilogue: BN(eval) + ReLU + sparsity mask, strided/offset store.
    if (!NTAIL || coB < COUT) {
        float scale = 1.0f, shift = 0.0f;
        if (DOBN) {
            const float g  = bn[0 * COUT + coB];
            const float bb = bn[1 * COUT + coB];
            const float rm = bn[2 * COUT + coB];
            const float rv = bn[3 * COUT + coB];
            scale = g * rsqrtf(rv + 1e-5f);
            shift = bb - rm * scale;
        }
        const int half8 = (lane >> 4) << 3;
#pragma unroll
        for (int r = 0; r < 8; ++r) {
            const int v0 = tileBase + r + half8;       // rows 0..15 tile
            const int v1 = v0 + 16;                    // rows 16..31 tile
            float y0 = acc0[r];
            float y1 = acc1[r];
            if (DOBN) {
                y0 = fmaxf(y0 * scale + shift, 0.0f);
                y1 = fmaxf(y1 * scale + shift, 0.0f);
            }
            y0 *= maskOut[v0];
            y1 *= maskOut[v1];
            out[(size_t)v0 * outStride + outOffset + coB] = y0;
            out[(size_t)v1 * outStride + outOffset + coB] = y1;
        }
    }
}

// ---------------------------------------------------------------------------
// Small helper kernels
// ---------------------------------------------------------------------------
__global__ void mask_f32_kernel(const int* __restrict__ m,
                                float* __restrict__ o, int n)
{
    const int i = blockIdx.x * blockDim.x + threadIdx.x;
    if (i < n) o[i] = (float)m[i];
}

// reduce_window max 3^3, stride 2, pad 1 (torchsparse stride-2 coords)
__global__ void pool_mask_kernel(const float* __restrict__ in,
                                 float* __restrict__ out, int Din, int Dout)
{
    const int v = blockIdx.x * blockDim.x + threadIdx.x;
    const int n = Dout * Dout * Dout;
    if (v >= n) return;
    const int x = v % Dout;
    const int y = (v / Dout) % Dout;
    const int z = v / (Dout * Dout);
    float m = 0.0f;
    for (int tz = 0; tz < 3; ++tz) {
        const int iz = z * 2 + tz - 1;
        if (iz < 0 || iz >= Din) continue;
        for (int ty = 0; ty < 3; ++ty) {
            const int iy = y * 2 + ty - 1;
            if (iy < 0 || iy >= Din) continue;
            for (int tx = 0; tx < 3; ++tx) {
                const int ix = x * 2 + tx - 1;
                if (ix < 0 || ix >= Din) continue;
                m = fmaxf(m, in[(iz * Din + iy) * Din + ix]);
            }
        }
    }
    out[v] = fminf(m, 1.0f);
}

__global__ void mask_mul_kernel(const float* __restrict__ f,
                                const float* __restrict__ m,
                                float* __restrict__ o, int C, int n)
{
    const int i = blockIdx.x * blockDim.x + threadIdx.x;
    if (i < n) o[i] = f[i] * m[i / C];
}

// skip-tensor drop into the upper channels of an in-place concat buffer
__global__ void concat_copy_kernel(const float* __restrict__ src,
                                   float* __restrict__ dst,
                                   int Csrc, int stride, int off, int n)
{
    const int i = blockIdx.x * blockDim.x + threadIdx.x;
    if (i < n) {
        const int v = i / Csrc;
        const int c = i - v * Csrc;
        dst[(size_t)v * stride + off + c] = src[i];
    }
}

// ---------------------------------------------------------------------------
// Host side
// ---------------------------------------------------------------------------
template <int CIN, int COUT, int DIN, int DOUT, int MODE, int DOBN>
static inline void launch_conv(const float* in, const float* w, const float* bn,
                               const float* maskOut, float* out,
                               int outStride, int outOffset, hipStream_t stream)
{
    constexpr int NW = (COUT + 15) / 16;
    dim3 grid((DOUT * DOUT * DOUT) / 32);
    dim3 block(32 * NW);
    conv_wmma_kernel<CIN, COUT, DIN, DOUT, MODE, DOBN>
        <<<grid, block, 0, stream>>>(in, w, bn, maskOut, out,
                                     outStride, outOffset);
}

static inline int cdiv(int a, int b) { return (a + b - 1) / b; }

extern "C" void kernel_launch(void* const* d_in, const int* in_sizes, int n_in,
                              void* d_out, int out_size, void* d_ws, size_t ws_size,
                              hipStream_t stream)
{
    (void)in_sizes; (void)n_in; (void)out_size; (void)ws_size;

    const float* feats = (const float*)d_in[0];
    const int*   maski = (const int*)d_in[1];
    const float* w0 = (const float*)d_in[2];
    const float* w1 = (const float*)d_in[3];
    const float* w2 = (const float*)d_in[4];
    const float* w3 = (const float*)d_in[5];
    const float* w4 = (const float*)d_in[6];
    const float* wt4 = (const float*)d_in[7];
    const float* wt3 = (const float*)d_in[8];
    const float* wt2 = (const float*)d_in[9];
    const float* wt1 = (const float*)d_in[10];
    const float* wout = (const float*)d_in[11];
    const float* bn0 = (const float*)d_in[12];
    const float* bn1 = (const float*)d_in[13];
    const float* bn2 = (const float*)d_in[14];
    const float* bn3 = (const float*)d_in[15];
    const float* bn4 = (const float*)d_in[16];
    const float* bnt4 = (const float*)d_in[17];
    const float* bnt3 = (const float*)d_in[18];
    const float* bnt2 = (const float*)d_in[19];
    const float* bnt1 = (const float*)d_in[20];

    const int V64 = 64 * 64 * 64, V32 = 32 * 32 * 32, V16 = 16 * 16 * 16;
    const int V8 = 8 * 8 * 8, V4 = 4 * 4 * 4;

    // carve fp32 workspace
    float* p = (float*)d_ws;
    float* m1  = p; p += V64;
    float* m2  = p; p += V32;
    float* m4  = p; p += V16;
    float* m8  = p; p += V8;
    float* m16 = p; p += V4;
    float* x0  = p; p += (size_t)V64 * 12;   // feats * m1
    float* s1  = p; p += (size_t)V64 * 8;
    float* s2  = p; p += (size_t)V32 * 16;
    float* s4  = p; p += (size_t)V16 * 32;
    float* s8  = p; p += (size_t)V8  * 64;
    float* s16 = p; p += (size_t)V4  * 128;
    float* cat8  = p; p += (size_t)V8  * 128; // [convT4 64 | s8 64]
    float* cat16 = p; p += (size_t)V16 * 96;  // [convT3 64 | s4 32]
    float* cat32 = p; p += (size_t)V32 * 64;  // [convT2 48 | s2 16]
    float* cat64 = p; p += (size_t)V64 * 40;  // [convT1 32 | s1  8]

    const int TB = 256;

    // mask pyramid
    mask_f32_kernel<<<cdiv(V64, TB), TB, 0, stream>>>(maski, m1, V64);
    pool_mask_kernel<<<cdiv(V32, TB), TB, 0, stream>>>(m1, m2, 64, 32);
    pool_mask_kernel<<<cdiv(V16, TB), TB, 0, stream>>>(m2, m4, 32, 16);
    pool_mask_kernel<<<cdiv(V8,  TB), TB, 0, stream>>>(m4, m8, 16, 8);
    pool_mask_kernel<<<cdiv(V4,  TB), TB, 0, stream>>>(m8, m16, 8, 4);

    // x0 = feats * m1
    mask_mul_kernel<<<cdiv(V64 * 12, TB), TB, 0, stream>>>(feats, m1, x0, 12, V64 * 12);

    // ------------------------- encoder -------------------------
    //            CIN COUT DIN DOUT MODE BN
    launch_conv<12,   8, 64, 64, 0, 1>(x0,  w0, bn0, m1,  s1,    8, 0, stream);
    launch_conv< 8,  16, 64, 32, 1, 1>(s1,  w1, bn1, m2,  s2,   16, 0, stream);
    launch_conv<16,  32, 32, 16, 1, 1>(s2,  w2, bn2, m4,  s4,   32, 0, stream);
    launch_conv<32,  64, 16,  8, 1, 1>(s4,  w3, bn3, m8,  s8,   64, 0, stream);
    launch_conv<64, 128,  8,  4, 1, 1>(s8,  w4, bn4, m16, s16, 128, 0, stream);

    // ------------------------- decoder -------------------------
    // convT4: s16 (4^3 x128) -> cat8[:, :64] (8^3), skip s8 -> cat8[:, 64:]
    launch_conv<128, 64,  4,  8, 2, 1>(s16, wt4, bnt4, m8, cat8, 128, 0, stream);
    concat_copy_kernel<<<cdiv(V8 * 64, TB), TB, 0, stream>>>(s8, cat8, 64, 128, 64, V8 * 64);

    // convT3: cat8 (8^3 x128) -> cat16[:, :64] (16^3), skip s4 -> cat16[:, 64:]
    launch_conv<128, 64,  8, 16, 2, 1>(cat8, wt3, bnt3, m4, cat16, 96, 0, stream);
    concat_copy_kernel<<<cdiv(V16 * 32, TB), TB, 0, stream>>>(s4, cat16, 32, 96, 64, V16 * 32);

    // convT2: cat16 (16^3 x96) -> cat32[:, :48] (32^3), skip s2 -> cat32[:, 48:]
    launch_conv< 96, 48, 16, 32, 2, 1>(cat16, wt2, bnt2, m2, cat32, 64, 0, stream);
    concat_copy_kernel<<<cdiv(V32 * 16, TB), TB, 0, stream>>>(s2, cat32, 16, 64, 48, V32 * 16);

    // convT1: cat32 (32^3 x64) -> cat64[:, :32] (64^3), skip s1 -> cat64[:, 32:]
    launch_conv< 64, 32, 32, 64, 2, 1>(cat32, wt1, bnt1, m1, cat64, 40, 0, stream);
    concat_copy_kernel<<<cdiv(V64 * 8, TB), TB, 0, stream>>>(s1, cat64, 8, 40, 32, V64 * 8);

    // final submanifold conv: cat64 (64^3 x40) -> out (64^3 x2), mask only
    launch_conv< 40,  2, 64, 64, 0, 0>(cat64, wout, nullptr, m1,
                                       (float*)d_out, 2, 0, stream);
}